// NIGConv_17051020165718
// MI455X (gfx1250) — compile-verified
//
#include <hip/hip_runtime.h>
#include <hip/hip_bf16.h>

typedef __attribute__((ext_vector_type(2))) float v2f;
typedef __attribute__((ext_vector_type(8))) float v8f;

#define N_NODES 100000
#define D 64
#define ROW_BLOCKS (N_NODES / 16)   // 6250, exact

// ---------------------------------------------------------------------------
// Kernel 1: zero the accumulator (d_out doubles as the neighbor-sum buffer)
// and the degree buffer in workspace.
// ---------------------------------------------------------------------------
__global__ void sage_zero_kernel(float* __restrict__ out, float* __restrict__ deg) {
    unsigned i = blockIdx.x * blockDim.x + threadIdx.x;
    if (i < (unsigned)(N_NODES * D)) out[i] = 0.0f;
    if (i < (unsigned)N_NODES)       deg[i] = 0.0f;
}

// ---------------------------------------------------------------------------
// Kernel 2: edge-parallel scatter. 16 lanes per edge, float4 slice per lane.
//   sum[dst] += feat[src] * w  ;  deg[dst] += 1
// ---------------------------------------------------------------------------
__global__ void sage_edge_scatter_kernel(const float* __restrict__ feat,
                                         const int* __restrict__ src,
                                         const int* __restrict__ dst,
                                         const float* __restrict__ ew,
                                         float* __restrict__ sum,
                                         float* __restrict__ deg,
                                         int n_edges) {
    unsigned t = blockIdx.x * blockDim.x + threadIdx.x;
    unsigned e = t >> 4;       // edge id
    unsigned q = t & 15u;      // quarter-row slot (4 floats each)
    if (e >= (unsigned)n_edges) return;

    int s = src[e];
    int d = dst[e];
    float w = ew[e];

    const float4* frow = (const float4*)(feat + (size_t)s * D);
    float4 v = frow[q];

    float* o = sum + (size_t)d * D + q * 4;
    atomicAdd(o + 0, v.x * w);
    atomicAdd(o + 1, v.y * w);
    atomicAdd(o + 2, v.z * w);
    atomicAdd(o + 3, v.w * w);
    if (q == 0) atomicAdd(deg + d, 1.0f);
}

// ---------------------------------------------------------------------------
// Kernel 3: fused WMMA GEMM.
// One wave32 owns a 16-row node block. A fragments (f32 16x16x4 layout:
// lanes 0-15 carry K={0,1} of row M=lane, lanes 16-31 carry K={2,3}) are
// preloaded into registers for both feat (self path) and sum*rdeg (neigh
// path); then 4 column tiles x 16 K-steps x 2 matrices of
// v_wmma_f32_16x16x4_f32 accumulate into one v8f, bias added, stored in place.
// ---------------------------------------------------------------------------
__global__ __launch_bounds__(256) void sage_wmma_gemm_kernel(
        const float* __restrict__ feat,
        const float* __restrict__ Wn,     // [D_OUT, D_IN] row-major
        const float* __restrict__ Ws,     // [D_OUT, D_IN] row-major
        const float* __restrict__ bias,
        const float* __restrict__ deg,
        float* __restrict__ out)          // in: sum accumulator, out: result
{
    unsigned wave = (blockIdx.x * blockDim.x + threadIdx.x) >> 5;
    unsigned lane = threadIdx.x & 31u;
    if (wave >= (unsigned)ROW_BLOCKS) return;   // whole-wave exit: EXEC stays all-1s

    const int m0   = (int)wave * 16;
    const int r    = (int)(lane & 15u);   // row within block (A), col within tile (B/C/D)
    const int half = (int)(lane >> 4);    // selects K pair {0,1} vs {2,3}

    const int row = m0 + r;
    const float rdeg = 1.0f / fmaxf(deg[row], 1.0f);

    // Preload all A fragments for this row block (32 + 32 VGPRs).
    v2f a_self[16];
    v2f a_neigh[16];
    const float* frow = feat + (size_t)row * D + half * 2;
    const float* srow = out  + (size_t)row * D + half * 2;
#pragma unroll
    for (int kk = 0; kk < 16; ++kk) {
        a_self[kk] = *(const v2f*)(frow + kk * 4);
        v2f sv     = *(const v2f*)(srow + kk * 4);
        a_neigh[kk] = sv * rdeg;
    }

#pragma unroll
    for (int jt = 0; jt < 4; ++jt) {
        const int j0 = jt * 16;
        v8f acc = {};
        const float* wsrow = Ws + (size_t)(j0 + r) * D + half * 2;
        const float* wnrow = Wn + (size_t)(j0 + r) * D + half * 2;
#pragma unroll
        for (int kk = 0; kk < 16; ++kk) {
            v2f bs = *(const v2f*)(wsrow + kk * 4);
            acc = __builtin_amdgcn_wmma_f32_16x16x4_f32(
                    false, a_self[kk], false, bs, (short)0, acc, false, false);
            v2f bn = *(const v2f*)(wnrow + kk * 4);
            acc = __builtin_amdgcn_wmma_f32_16x16x4_f32(
                    false, a_neigh[kk], false, bn, (short)0, acc, false, false);
        }
        const float bj = bias[j0 + r];
        // C/D layout: VGPR rr, lane -> row m0 + rr + half*8, col j0 + r
#pragma unroll
        for (int rr = 0; rr < 8; ++rr) {
            out[(size_t)(m0 + rr + half * 8) * D + j0 + r] = acc[rr] + bj;
        }
    }
}

// ---------------------------------------------------------------------------
extern "C" void kernel_launch(void* const* d_in, const int* in_sizes, int n_in,
                              void* d_out, int out_size, void* d_ws, size_t ws_size,
                              hipStream_t stream) {
    const float* feat = (const float*)d_in[0];
    const int*   src  = (const int*)  d_in[1];
    const int*   dst  = (const int*)  d_in[2];
    const float* ew   = (const float*)d_in[3];
    const float* Wn   = (const float*)d_in[4];
    const float* Ws   = (const float*)d_in[5];
    const float* bias = (const float*)d_in[6];

    float* out = (float*)d_out;
    float* deg = (float*)d_ws;              // N_NODES floats (400 KB) of scratch
    const int n_edges = in_sizes[1];

    // 1) zero accumulator + degrees
    {
        int total = N_NODES * D;
        int blocks = (total + 255) / 256;
        sage_zero_kernel<<<blocks, 256, 0, stream>>>(out, deg);
    }
    // 2) edge scatter (16 lanes / edge)
    {
        long long threads = (long long)n_edges * 16;
        int blocks = (int)((threads + 255) / 256);
        sage_edge_scatter_kernel<<<blocks, 256, 0, stream>>>(feat, src, dst, ew,
                                                             out, deg, n_edges);
    }
    // 3) fused WMMA GEMM, one wave per 16-row block
    {
        int waves = ROW_BLOCKS;
        int blocks = (waves + 7) / 8;       // 8 waves per 256-thread block
        sage_wmma_gemm_kernel<<<blocks, 256, 0, stream>>>(feat, Wn, Ws, bias,
                                                          deg, out);
    }
}